// MPNN_79628693668165
// MI455X (gfx1250) — compile-verified
//
#include <hip/hip_runtime.h>

typedef __attribute__((ext_vector_type(2))) float v2f;
typedef __attribute__((ext_vector_type(8))) float v8f;

#define N_NODES 10000
#define F_IN    256
#define HIDDEN  512
#define C_OUT   128

// ---------------------------------------------------------------- zero fill
__global__ void zero_f32(float* __restrict__ p, long n) {
  long i = (long)blockIdx.x * blockDim.x + threadIdx.x;
  if (i < n) p[i] = 0.0f;
}

// ------------------------------------------------- edge scatter-add (float4)
// one thread handles 4 consecutive features of one edge
template <int F>
__global__ void scatter_add4(const float* __restrict__ x,
                             const int* __restrict__ src,
                             const int* __restrict__ dst,
                             float* __restrict__ agg,
                             int E) {
  constexpr int fdiv4 = F >> 2;
  long gid = (long)blockIdx.x * blockDim.x + threadIdx.x;
  long total = (long)E * fdiv4;
  if (gid >= total) return;
  int e = (int)(gid / fdiv4);
  int f = (int)(gid % fdiv4) << 2;
  int s = src[e];
  int d = dst[e];
  const float4 v = *(const float4*)(x + (long)s * F + f);
  float* p = agg + (long)d * F + f;
  atomicAdd(p + 0, v.x);
  atomicAdd(p + 1, v.y);
  atomicAdd(p + 2, v.z);
  atomicAdd(p + 3, v.w);
}

// -------------------------------------------------------- dual-GEMM via WMMA
// out = maybe_relu( A1*B1 + A2*B2 + bias )
// A1,A2: [M x K] row-major;  B1,B2: [K x NC] row-major;  out: [M x NC]
// one wave computes one 16x16 tile with v_wmma_f32_16x16x4_f32 over K.
// K, NC compile-time -> B loads use literal IOFFSETs (max 512KB < 2^23),
// K-loop unrolls. M=10000, K,NC multiples of 16 -> no tails, EXEC all-1s.
template <int K, int NC, int RELU>
__global__ void sage_gemm_wmma(const float* __restrict__ A1,
                               const float* __restrict__ B1,
                               const float* __restrict__ A2,
                               const float* __restrict__ B2,
                               const float* __restrict__ bias,
                               float* __restrict__ out) {
  const int lane = threadIdx.x & 31;
  const int wave = threadIdx.x >> 5;
  const int col0 = (blockIdx.x * 8 + wave) * 16;   // 8 waves/block, adjacent N-tiles
  const int row0 = blockIdx.y * 16;
  const int half = lane >> 4;                      // 0: lanes 0-15, 1: lanes 16-31
  const int lm   = lane & 15;

  // A(16x4) layout: lane holds A[m=lm, k = 2*half + v], v in {0,1}  (contiguous)
  const float* a1p = A1 + (long)(row0 + lm) * K + 2 * half;
  const float* a2p = A2 + (long)(row0 + lm) * K + 2 * half;
  // B(4x16) layout: vgpr v holds B[k = v + 2*half, n = lm]
  const float* b1p = B1 + (long)(2 * half) * NC + col0 + lm;
  const float* b2p = B2 + (long)(2 * half) * NC + col0 + lm;

  v8f c = {};
#pragma unroll 8
  for (int kb = 0; kb < K; kb += 4) {
    v2f a1 = *(const v2f*)(a1p + kb);
    v2f a2 = *(const v2f*)(a2p + kb);
    v2f b1;
    b1.x = b1p[kb * NC];
    b1.y = b1p[(kb + 1) * NC];
    v2f b2;
    b2.x = b2p[kb * NC];
    b2.y = b2p[(kb + 1) * NC];
    c = __builtin_amdgcn_wmma_f32_16x16x4_f32(false, a1, false, b1,
                                              (short)0, c, false, false);
    c = __builtin_amdgcn_wmma_f32_16x16x4_f32(false, a2, false, b2,
                                              (short)0, c, false, false);
  }

  const float bb = bias[col0 + lm];
  // C(16x16) layout: vgpr v -> row m = v + 8*half, col n = lm
  float* op = out + (long)(row0 + 8 * half) * NC + col0 + lm;
#pragma unroll
  for (int v = 0; v < 8; ++v) {
    float r = c[v] + bb;
    if (RELU) r = fmaxf(r, 0.0f);
    op[v * NC] = r;
  }
}

// --------------------------------------------------------------------- launch
extern "C" void kernel_launch(void* const* d_in, const int* in_sizes, int n_in,
                              void* d_out, int out_size, void* d_ws, size_t ws_size,
                              hipStream_t stream) {
  const float* x   = (const float*)d_in[0];
  const int*   ei  = (const int*)d_in[1];      // [2, E] flattened
  const float* W1l = (const float*)d_in[2];    // [F_IN, HID]
  const float* b1  = (const float*)d_in[3];    // [HID]
  const float* W1r = (const float*)d_in[4];    // [F_IN, HID]
  const float* W2l = (const float*)d_in[5];    // [HID, C_OUT]
  const float* b2  = (const float*)d_in[6];    // [C_OUT]
  const float* W2r = (const float*)d_in[7];    // [HID, C_OUT]
  float* out = (float*)d_out;

  const int E = in_sizes[1] / 2;
  const int* src = ei;
  const int* dst = ei + E;

  float* agg1 = (float*)d_ws;                         // N * F_IN
  float* h    = agg1 + (long)N_NODES * F_IN;          // N * HID
  float* agg2 = h    + (long)N_NODES * HIDDEN;        // N * HID

  // ---- layer 1: agg1 = segment_sum(x[src], dst)
  {
    long n = (long)N_NODES * F_IN;
    zero_f32<<<(unsigned)((n + 255) / 256), 256, 0, stream>>>(agg1, n);
    long total = (long)E * (F_IN / 4);
    scatter_add4<F_IN><<<(unsigned)((total + 255) / 256), 256, 0, stream>>>(
        x, src, dst, agg1, E);
  }
  // ---- layer 1 GEMM: h = relu(agg1@W1l + x@W1r + b1)
  {
    dim3 grid(HIDDEN / 128, N_NODES / 16);   // (4, 625), 8 waves per block
    sage_gemm_wmma<F_IN, HIDDEN, 1><<<grid, 256, 0, stream>>>(
        agg1, W1l, x, W1r, b1, h);
  }
  // ---- layer 2: agg2 = segment_sum(h[src], dst)
  {
    long n = (long)N_NODES * HIDDEN;
    zero_f32<<<(unsigned)((n + 255) / 256), 256, 0, stream>>>(agg2, n);
    long total = (long)E * (HIDDEN / 4);
    scatter_add4<HIDDEN><<<(unsigned)((total + 255) / 256), 256, 0, stream>>>(
        h, src, dst, agg2, E);
  }
  // ---- layer 2 GEMM: out = agg2@W2l + h@W2r + b2
  {
    dim3 grid(C_OUT / 128, N_NODES / 16);    // (1, 625), 8 waves per block
    sage_gemm_wmma<HIDDEN, C_OUT, 0><<<grid, 256, 0, stream>>>(
        agg2, W2l, h, W2r, b2, out);
  }
}